// WeightedProcrustes_61151744360524
// MI455X (gfx1250) — compile-verified
//
#include <hip/hip_runtime.h>
#include <math.h>

typedef __attribute__((ext_vector_type(2))) float v2f;
typedef __attribute__((ext_vector_type(8))) float v8f;

#define CHUNKS 64
#define BLOCK  256
#define NACC   23
#define PCOLS  32
#define NWAVES (BLOCK / 32)

// ---------------- Stage 1: streaming moment accumulation ----------------
// acc layout (per batch):
//  0        : sum w                     (thresholded raw weights)
//  1..3     : sum w * s
//  4..6     : sum w * r
//  7        : sum wv                    (wv = w * valid(src != 0))
//  8..10    : sum wv * s
//  11..13   : sum wv * r
//  14..22   : sum wv * s_i * r_j   (i*3+j)
__global__ void wp_stage1(const float* __restrict__ src,
                          const float* __restrict__ ref,
                          const float* __restrict__ wts,
                          float* __restrict__ P, int N) {
  const int chunk = blockIdx.x;
  const int b     = blockIdx.y;
  const long long baseP = (long long)b * (long long)N;

  const int per   = (N + CHUNKS - 1) / CHUNKS;
  const int start = chunk * per;
  const int end   = (start + per < N) ? (start + per) : N;

  float acc[NACC];
#pragma unroll
  for (int i = 0; i < NACC; ++i) acc[i] = 0.0f;

  for (int n = start + (int)threadIdx.x; n < end; n += BLOCK) {
    const long long p = baseP + n;
    float w = wts[p];
    w = (w < 0.0f) ? 0.0f : w;  // WEIGHT_THRESH = 0.0

    const float sx = src[3 * p + 0], sy = src[3 * p + 1], sz = src[3 * p + 2];
    const float rx = ref[3 * p + 0], ry = ref[3 * p + 1], rz = ref[3 * p + 2];

    const bool valid = !((sx == 0.0f) && (sy == 0.0f) && (sz == 0.0f));
    const float wv = valid ? w : 0.0f;

    acc[0] += w;
    acc[1] += w * sx;  acc[2] += w * sy;  acc[3] += w * sz;
    acc[4] += w * rx;  acc[5] += w * ry;  acc[6] += w * rz;
    acc[7] += wv;
    acc[8]  += wv * sx; acc[9]  += wv * sy; acc[10] += wv * sz;
    acc[11] += wv * rx; acc[12] += wv * ry; acc[13] += wv * rz;
    const float wsx = wv * sx, wsy = wv * sy, wsz = wv * sz;
    acc[14] += wsx * rx; acc[15] += wsx * ry; acc[16] += wsx * rz;
    acc[17] += wsy * rx; acc[18] += wsy * ry; acc[19] += wsy * rz;
    acc[20] += wsz * rx; acc[21] += wsz * ry; acc[22] += wsz * rz;
  }

  // wave32 reduction
#pragma unroll
  for (int i = 0; i < NACC; ++i) {
    float v = acc[i];
#pragma unroll
    for (int off = 16; off > 0; off >>= 1) v += __shfl_xor(v, off, 32);
    acc[i] = v;
  }

  __shared__ float lds[NWAVES][NACC];
  const int lane = threadIdx.x & 31;
  const int wid  = threadIdx.x >> 5;
  if (lane == 0) {
#pragma unroll
    for (int i = 0; i < NACC; ++i) lds[wid][i] = acc[i];
  }
  __syncthreads();

  if (threadIdx.x < PCOLS) {
    float s = 0.0f;
    if (threadIdx.x < NACC) {
#pragma unroll
      for (int w8 = 0; w8 < NWAVES; ++w8) s += lds[w8][threadIdx.x];
    }
    // padded columns [23,32) written as 0 so WMMA tiles are clean
    P[((long long)b * CHUNKS + chunk) * PCOLS + threadIdx.x] = s;
  }
}

// ---------------- small helpers for the 3x3 solve ----------------
__device__ inline float det3(const float M[3][3]) {
  return M[0][0] * (M[1][1] * M[2][2] - M[1][2] * M[2][1]) -
         M[0][1] * (M[1][0] * M[2][2] - M[1][2] * M[2][0]) +
         M[0][2] * (M[1][0] * M[2][1] - M[1][1] * M[2][0]);
}

__device__ void jacobi3(float A[3][3], float V[3][3]) {
  for (int i = 0; i < 3; ++i)
    for (int j = 0; j < 3; ++j) V[i][j] = (i == j) ? 1.0f : 0.0f;
  const int PQ[3][2] = {{0, 1}, {0, 2}, {1, 2}};
  for (int sweep = 0; sweep < 12; ++sweep) {
    for (int r = 0; r < 3; ++r) {
      const int p = PQ[r][0], q = PQ[r][1];
      const float apq = A[p][q];
      if (fabsf(apq) < 1e-30f) continue;
      const float tau = (A[q][q] - A[p][p]) / (2.0f * apq);
      const float t = (tau >= 0.0f) ? 1.0f / (tau + sqrtf(1.0f + tau * tau))
                                    : 1.0f / (tau - sqrtf(1.0f + tau * tau));
      const float c = 1.0f / sqrtf(1.0f + t * t);
      const float s = t * c;
      for (int k = 0; k < 3; ++k) {           // A <- A G
        const float akp = A[k][p], akq = A[k][q];
        A[k][p] = c * akp - s * akq;
        A[k][q] = s * akp + c * akq;
      }
      for (int k = 0; k < 3; ++k) {           // A <- G^T A
        const float apk = A[p][k], aqk = A[q][k];
        A[p][k] = c * apk - s * aqk;
        A[q][k] = s * apk + c * aqk;
      }
      for (int k = 0; k < 3; ++k) {           // V <- V G
        const float vkp = V[k][p], vkq = V[k][q];
        V[k][p] = c * vkp - s * vkq;
        V[k][q] = s * vkp + c * vkq;
      }
    }
  }
}

// ---------------- Stage 2: WMMA cross-chunk reduction + 3x3 SVD ----------------
__global__ void wp_stage2(const float* __restrict__ P, float* __restrict__ out) {
  const int b    = blockIdx.x;
  const int lane = threadIdx.x;        // blockDim.x == 32, one full wave
  const float* Pb = P + (size_t)b * CHUNKS * PCOLS;

  const int col  = lane & 15;
  const int rsel = (lane >> 4) * 2;    // which pair of K-rows this half-wave holds

  v2f ones; ones.x = 1.0f; ones.y = 1.0f;   // A = 16x4 all-ones
  v8f cl = {}; v8f ch = {};

#pragma unroll
  for (int base = 0; base < CHUNKS; base += 4) {
    const float* r0 = Pb + (base + rsel + 0) * PCOLS;
    const float* r1 = Pb + (base + rsel + 1) * PCOLS;
    v2f blo, bhi;
    blo.x = r0[col];      blo.y = r1[col];
    bhi.x = r0[col + 16]; bhi.y = r1[col + 16];
    // C[m][j] += sum_k B[k][j]  (invariant to K-row lane placement since A==1)
    cl = __builtin_amdgcn_wmma_f32_16x16x4_f32(false, ones, false, blo,
                                               (short)0, cl, false, false);
    ch = __builtin_amdgcn_wmma_f32_16x16x4_f32(false, ones, false, bhi,
                                               (short)0, ch, false, false);
  }

  // D row M=0 lives in VGPR0, lanes 0..15 (documented C/D layout)
  __shared__ float tot[PCOLS];
  if (lane < 16) {
    tot[lane]      = cl[0];
    tot[lane + 16] = ch[0];
  }
  __syncthreads();

  if (lane != 0) return;

  const float f = 1.0f / (tot[0] + 1e-5f);   // EPS
  float sc[3]  = {tot[1] * f, tot[2] * f, tot[3] * f};    // src centroid
  float rc[3]  = {tot[4] * f, tot[5] * f, tot[6] * f};    // ref centroid
  const float Swv = tot[7] * f;
  float Ssv[3] = {tot[8] * f, tot[9] * f, tot[10] * f};
  float Srv[3] = {tot[11] * f, tot[12] * f, tot[13] * f};

  float H[3][3];
  for (int i = 0; i < 3; ++i)
    for (int j = 0; j < 3; ++j)
      H[i][j] = tot[14 + i * 3 + j] * f - sc[i] * Srv[j] - Ssv[i] * rc[j] +
                Swv * sc[i] * rc[j];

  // HtH and its eigen-decomposition -> V, lambda
  float A[3][3];
  for (int i = 0; i < 3; ++i)
    for (int j = 0; j < 3; ++j)
      A[i][j] = H[0][i] * H[0][j] + H[1][i] * H[1][j] + H[2][i] * H[2][j];
  float V[3][3];
  jacobi3(A, V);
  float lam[3] = {A[0][0], A[1][1], A[2][2]};

  // sort eigenpairs descending
  int idx[3] = {0, 1, 2};
  for (int i = 0; i < 2; ++i)
    for (int j = i + 1; j < 3; ++j)
      if (lam[idx[j]] > lam[idx[i]]) { int t = idx[i]; idx[i] = idx[j]; idx[j] = t; }
  float Vs[3][3];
  for (int i = 0; i < 3; ++i)
    for (int k = 0; k < 3; ++k) Vs[i][k] = V[i][idx[k]];

  // U columns = H * v_k / sigma_k
  float BH[3][3], U[3][3];
  for (int k = 0; k < 3; ++k)
    for (int i = 0; i < 3; ++i)
      BH[i][k] = H[i][0] * Vs[0][k] + H[i][1] * Vs[1][k] + H[i][2] * Vs[2][k];
  for (int k = 0; k < 3; ++k) {
    const float nx = BH[0][k], ny = BH[1][k], nz = BH[2][k];
    const float n = sqrtf(nx * nx + ny * ny + nz * nz);
    if (n > 1e-20f) {
      const float inv = 1.0f / n;
      U[0][k] = nx * inv; U[1][k] = ny * inv; U[2][k] = nz * inv;
    } else if (k == 2) {
      float cx = U[1][0] * U[2][1] - U[2][0] * U[1][1];
      float cy = U[2][0] * U[0][1] - U[0][0] * U[2][1];
      float cz = U[0][0] * U[1][1] - U[1][0] * U[0][1];
      const float cn = sqrtf(cx * cx + cy * cy + cz * cz) + 1e-30f;
      U[0][2] = cx / cn; U[1][2] = cy / cn; U[2][2] = cz / cn;
    } else {
      U[0][k] = (k == 0) ? 1.0f : 0.0f;
      U[1][k] = (k == 1) ? 1.0f : 0.0f;
      U[2][k] = 0.0f;
    }
  }

  const float d = (det3(Vs) * det3(U) < 0.0f) ? -1.0f : 1.0f;

  // R = Vs * diag(1,1,d) * U^T
  float R[3][3];
  for (int i = 0; i < 3; ++i)
    for (int j = 0; j < 3; ++j)
      R[i][j] = Vs[i][0] * U[j][0] + Vs[i][1] * U[j][1] + d * Vs[i][2] * U[j][2];

  float tv[3];
  for (int i = 0; i < 3; ++i)
    tv[i] = rc[i] - (R[i][0] * sc[0] + R[i][1] * sc[1] + R[i][2] * sc[2]);

  float* T = out + (size_t)b * 16;
  for (int i = 0; i < 3; ++i) {
    T[i * 4 + 0] = R[i][0];
    T[i * 4 + 1] = R[i][1];
    T[i * 4 + 2] = R[i][2];
    T[i * 4 + 3] = tv[i];
  }
  T[12] = 0.0f; T[13] = 0.0f; T[14] = 0.0f; T[15] = 1.0f;
}

extern "C" void kernel_launch(void* const* d_in, const int* in_sizes, int n_in,
                              void* d_out, int out_size, void* d_ws, size_t ws_size,
                              hipStream_t stream) {
  const float* src = (const float*)d_in[0];
  const float* ref = (const float*)d_in[1];
  const float* wts = (const float*)d_in[2];
  float* out = (float*)d_out;
  float* P   = (float*)d_ws;     // needs B * CHUNKS * PCOLS * 4 bytes (512 KB @ B=64)

  const int B = out_size / 16;          // (B,4,4) output
  const int N = in_sizes[2] / B;        // weights are (B,N)

  dim3 g1(CHUNKS, B);
  wp_stage1<<<g1, BLOCK, 0, stream>>>(src, ref, wts, P, N);
  wp_stage2<<<B, 32, 0, stream>>>(P, out);
}